// SelectiveStateSpace_46179488367075
// MI455X (gfx1250) — compile-verified
//
#include <hip/hip_runtime.h>
#include <hip/hip_bf16.h>

// ---------------------------------------------------------------------------
// Mamba selective-SSM forward on gfx1250.
// Dims (fixed by reference): B=1, L=1024, DM=1024, DIN=2048, N=16, K=4, R=64
// ---------------------------------------------------------------------------
#define L_SEQ 1024
#define DM    1024
#define DIN   2048
#define NST   16
#define KCONV 4
#define RRANK 64

typedef __attribute__((ext_vector_type(2))) float v2f;
typedef __attribute__((ext_vector_type(8))) float v8f;
typedef unsigned int u32;
typedef unsigned long long u64;
typedef __attribute__((ext_vector_type(4))) u32 u32x4;
typedef __attribute__((ext_vector_type(4))) int i32x4;
typedef __attribute__((ext_vector_type(8))) int i32x8;

#ifndef __has_builtin
#define __has_builtin(x) 0
#endif
#if __has_builtin(__builtin_amdgcn_tensor_load_to_lds)
#define USE_TDM 1
#else
#define USE_TDM 0
#endif

__device__ __forceinline__ float softplus_f(float x) {
    return (x > 20.f) ? x : log1pf(__expf(x));
}
__device__ __forceinline__ float silu_f(float x) {
    return x / (1.f + __expf(-x));
}

#define KC 16   // K-chunk staged in LDS per iteration

#if USE_TDM
// ---------------------------------------------------------------------------
// Issue one TDM 2D tile load: tileRows x tileCols f32 elements from a
// row-major tensor (row stride = strideElems) into LDS at ldsAddr, with a
// 1-DWORD pad inserted after every 16 DWORDs (reproduces [rows][17] layout).
// Rows beyond tensorRows and cols beyond tensorCols are zero-filled by HW.
// D# packing per CDNA5 ISA §8.3/8.4 (group0 128b, group1 256b).
// ---------------------------------------------------------------------------
__device__ __forceinline__ void tdm_load_tile(
    u32 ldsAddr, const float* gtile, u64 strideElems,
    u32 tensorCols, u32 tensorRows, u32 tileCols, u32 tileRows) {
    const u64 gaddr = (u64)(uintptr_t)gtile;
    u32x4 g0;
    g0[0] = 1u;                                      // count=1, no gather
    g0[1] = ldsAddr;                                 // lds_addr
    g0[2] = (u32)(gaddr & 0xFFFFFFFFull);            // global_addr[31:0]
    g0[3] = (u32)((gaddr >> 32) & 0x01FFFFFFu) | 0x80000000u;  // [56:32]|type=2

    // group1 bits: data_size=2(4B)@16, pad_enable@20, pad_interval=3(16DW)@22,
    // pad_amount=0(1DW)@25; tensor_dim0@[79:48]; tensor_dim1@[111:80];
    // tile_dim0@[127:112]; tile_dim1@[143:128]; tensor_dim0_stride@[207:160]
    u64 q0 = (2ull << 16) | (1ull << 20) | (3ull << 22);
    q0 |= ((u64)(tensorCols & 0xFFFFu)) << 48;
    u64 q1 = ((u64)(tensorCols >> 16))
           | ((u64)tensorRows << 16)
           | ((u64)(tileCols & 0xFFFFu) << 48);
    u64 q2 = ((u64)(tileRows & 0xFFFFu))
           | ((strideElems & 0xFFFFFFFFull) << 32);
    u64 q3 = (strideElems >> 32) & 0xFFFFull;
    i32x8 g1;
    g1[0] = (int)(u32)q0; g1[1] = (int)(u32)(q0 >> 32);
    g1[2] = (int)(u32)q1; g1[3] = (int)(u32)(q1 >> 32);
    g1[4] = (int)(u32)q2; g1[5] = (int)(u32)(q2 >> 32);
    g1[6] = (int)(u32)q3; g1[7] = (int)(u32)(q3 >> 32);
    i32x4 z4 = {0, 0, 0, 0};
#if __has_include(<hip/amd_detail/amd_gfx1250_TDM.h>)
    i32x8 z8 = {0, 0, 0, 0, 0, 0, 0, 0};
    __builtin_amdgcn_tensor_load_to_lds(g0, g1, z4, z4, z8, 0);
#else
    __builtin_amdgcn_tensor_load_to_lds(g0, g1, z4, z4, 0);
#endif
}
#endif  // USE_TDM

// ---------------------------------------------------------------------------
// LDS-staged, double-buffered f32 WMMA GEMM:
//   C[M,N] = A[M,K] * B^T, B row-major [N,K]
// Block = 256 threads = 8 waves; block tile 64(M) x 64(N); each wave owns a
// 16x32 strip (two accumulators).  K chunks of 16 staged into LDS by the TDM
// (wave 0 issues tensor_load_to_lds for chunk i+1 while all waves run the
// 8 WMMAs of chunk i; s_wait_tensorcnt + split barrier publishes the data).
// mode==1: fused bias + softplus epilogue.
// ---------------------------------------------------------------------------
__global__ __launch_bounds__(256)
void wmma_gemm_f32(const float* __restrict__ A, const float* __restrict__ Bm,
                   float* __restrict__ C, int M, int Ncols, int K,
                   int lda, int ldb, int ldc,
                   const float* __restrict__ bias, int mode) {
    __shared__ float lA[2][64][KC + 1];
    __shared__ float lB[2][64][KC + 1];

    const int tid    = threadIdx.x;
    const int lane   = tid & 31;
    const int waveId = tid >> 5;
    const int mloc = (waveId & 3) * 16;   // 4 row strips of 16
    const int nloc = (waveId >> 2) * 32;  // 2 col strips of 32
    const int m0   = blockIdx.x * 64;
    const int n0   = blockIdx.y * 64;

    const int l15 = lane & 15;
    const int kb  = (lane >> 4) * 2;      // K sub-offset per half-wave
    const int nChunks = K / KC;

    v8f acc0 = {};
    v8f acc1 = {};

#if USE_TDM
    const u32 bRows = (u32)(Ncols - n0 < 64 ? Ncols - n0 : 64);
    if (waveId == 0) {   // prologue: DMA chunk 0 into buffer 0
        tdm_load_tile((u32)(uintptr_t)&lA[0][0][0], A + (size_t)m0 * lda,
                      (u64)lda, (u32)K, 64u, KC, 64u);
        tdm_load_tile((u32)(uintptr_t)&lB[0][0][0], Bm + (size_t)n0 * ldb,
                      (u64)ldb, (u32)K, bRows, KC, 64u);
        __builtin_amdgcn_s_wait_tensorcnt(0);
    }
    __syncthreads();

    for (int i = 0; i < nChunks; ++i) {
        const int buf = i & 1;
        if (waveId == 0 && i + 1 < nChunks) {   // DMA next chunk into buf^1
            const int kn = (i + 1) * KC;
            tdm_load_tile((u32)(uintptr_t)&lA[buf ^ 1][0][0],
                          A + (size_t)m0 * lda + kn, (u64)lda,
                          (u32)(K - kn), 64u, KC, 64u);
            tdm_load_tile((u32)(uintptr_t)&lB[buf ^ 1][0][0],
                          Bm + (size_t)n0 * ldb + kn, (u64)ldb,
                          (u32)(K - kn), bRows, KC, 64u);
        }
#pragma unroll
        for (int kk = 0; kk < KC; kk += 4) {
            v2f a, b0, b1;
            a.x  = lA[buf][mloc + l15][kk + kb];
            a.y  = lA[buf][mloc + l15][kk + kb + 1];
            b0.x = lB[buf][nloc + l15][kk + kb];
            b0.y = lB[buf][nloc + l15][kk + kb + 1];
            b1.x = lB[buf][nloc + 16 + l15][kk + kb];
            b1.y = lB[buf][nloc + 16 + l15][kk + kb + 1];
            acc0 = __builtin_amdgcn_wmma_f32_16x16x4_f32(
                false, a, false, b0, (short)0, acc0, false, false);
            acc1 = __builtin_amdgcn_wmma_f32_16x16x4_f32(
                false, a, false, b1, (short)0, acc1, false, false);
        }
        if (i + 1 < nChunks) {
            if (waveId == 0) __builtin_amdgcn_s_wait_tensorcnt(0);
            __syncthreads();
        }
    }
#else
    // Fallback: cooperative staged loads (thread t -> 4 floats of 64x16 tile)
    const int srow = tid >> 2;
    const int scol = (tid & 3) * 4;
    const int aRow = m0 + srow;
    const int bRow = n0 + srow;
    const bool bValid = (bRow < Ncols);
    for (int i = 0; i < nChunks; ++i) {
        const int k0  = i * KC;
        const int buf = i & 1;
        float4 av = *(const float4*)(A + (size_t)aRow * lda + k0 + scol);
        float4 bv = bValid ? *(const float4*)(Bm + (size_t)bRow * ldb + k0 + scol)
                           : make_float4(0.f, 0.f, 0.f, 0.f);
        lA[buf][srow][scol + 0] = av.x;  lA[buf][srow][scol + 1] = av.y;
        lA[buf][srow][scol + 2] = av.z;  lA[buf][srow][scol + 3] = av.w;
        lB[buf][srow][scol + 0] = bv.x;  lB[buf][srow][scol + 1] = bv.y;
        lB[buf][srow][scol + 2] = bv.z;  lB[buf][srow][scol + 3] = bv.w;
        __syncthreads();
#pragma unroll
        for (int kk = 0; kk < KC; kk += 4) {
            v2f a, b0, b1;
            a.x  = lA[buf][mloc + l15][kk + kb];
            a.y  = lA[buf][mloc + l15][kk + kb + 1];
            b0.x = lB[buf][nloc + l15][kk + kb];
            b0.y = lB[buf][nloc + l15][kk + kb + 1];
            b1.x = lB[buf][nloc + 16 + l15][kk + kb];
            b1.y = lB[buf][nloc + 16 + l15][kk + kb + 1];
            acc0 = __builtin_amdgcn_wmma_f32_16x16x4_f32(
                false, a, false, b0, (short)0, acc0, false, false);
            acc1 = __builtin_amdgcn_wmma_f32_16x16x4_f32(
                false, a, false, b1, (short)0, acc1, false, false);
        }
        __syncthreads();
    }
#endif

    // ---- epilogue: C/D layout VGPR i -> row +i + 8*(lane>>4) -----------
    const int rbase = m0 + mloc + ((lane >> 4) << 3);
    const int nc0 = n0 + nloc + l15;
    const int nc1 = nc0 + 16;
#pragma unroll
    for (int i = 0; i < 8; ++i) {
        const int row = rbase + i;
        if (nc0 < Ncols) {
            float v = acc0[i];
            if (mode == 1) v = softplus_f(v + bias[nc0]);
            C[(size_t)row * ldc + nc0] = v;
        }
        if (nc1 < Ncols) {
            float v = acc1[i];
            if (mode == 1) v = softplus_f(v + bias[nc1]);
            C[(size_t)row * ldc + nc1] = v;
        }
    }
}

// ---------------------------------------------------------------------------
// Causal depthwise conv1d (K=4, left pad 3) + bias + silu.
// ---------------------------------------------------------------------------
__global__ __launch_bounds__(256)
void conv_silu_kernel(const float* __restrict__ xz,
                      const float* __restrict__ conv_w,
                      const float* __restrict__ conv_b,
                      float* __restrict__ xc) {
    const int d = blockIdx.x * 256 + threadIdx.x;
    const int l = blockIdx.y;
    float s = conv_b[d];
#pragma unroll
    for (int j = 0; j < KCONV; ++j) {
        const int tl = l - (KCONV - 1) + j;
        if (tl >= 0) s += xz[(size_t)tl * (2 * DIN) + d] * conv_w[d * KCONV + j];
    }
    xc[(size_t)l * DIN + d] = silu_f(s);
}

// ---------------------------------------------------------------------------
// Fused selective scan + C-contraction + skip + silu(z) gate.
// One block = 16 channels x 16 states; wave32 shuffle reduction over n.
// ---------------------------------------------------------------------------
__global__ __launch_bounds__(256)
void scan_kernel(const float* __restrict__ dtm,
                 const float* __restrict__ xdbl,
                 const float* __restrict__ xc,
                 const float* __restrict__ xz,
                 const float* __restrict__ A_log,
                 const float* __restrict__ D_param,
                 float* __restrict__ y) {
    const int n = threadIdx.x & 15;
    const int d = blockIdx.x * 16 + (threadIdx.x >> 4);

    const float Ad = -__expf(A_log[d * NST + n]);
    const float Dv = D_param[d];
    float h = 0.f;

    for (int t = 0; t < L_SEQ; ++t) {
        const float dtv = dtm[(size_t)t * DIN + d];
        const float xcv = xc[(size_t)t * DIN + d];
        const float bn  = xdbl[(size_t)t * (RRANK + 2 * NST) + RRANK + n];
        const float cn  = xdbl[(size_t)t * (RRANK + 2 * NST) + RRANK + NST + n];
        h = __expf(dtv * Ad) * h + (dtv * bn) * xcv;
        float p = h * cn;
        p += __shfl_xor(p, 8, 16);
        p += __shfl_xor(p, 4, 16);
        p += __shfl_xor(p, 2, 16);
        p += __shfl_xor(p, 1, 16);
        if (n == 0) {
            const float zv = xz[(size_t)t * (2 * DIN) + DIN + d];
            y[(size_t)t * DIN + d] = (p + Dv * xcv) * silu_f(zv);
        }
    }
}

// ---------------------------------------------------------------------------
// Launcher.
// ---------------------------------------------------------------------------
extern "C" void kernel_launch(void* const* d_in, const int* in_sizes, int n_in,
                              void* d_out, int out_size, void* d_ws, size_t ws_size,
                              hipStream_t stream) {
    const float* x       = (const float*)d_in[0];
    const float* W_in    = (const float*)d_in[1];
    const float* conv_w  = (const float*)d_in[2];
    const float* conv_b  = (const float*)d_in[3];
    const float* W_x     = (const float*)d_in[4];
    const float* W_dt    = (const float*)d_in[5];
    const float* b_dt    = (const float*)d_in[6];
    const float* A_log   = (const float*)d_in[7];
    const float* D_param = (const float*)d_in[8];
    const float* W_out   = (const float*)d_in[9];
    float* out = (float*)d_out;

    float* ws    = (float*)d_ws;
    float* xz    = ws;                                   // L * 2*DIN
    float* xc    = xz    + (size_t)L_SEQ * 2 * DIN;      // L * DIN
    float* x_dbl = xc    + (size_t)L_SEQ * DIN;          // L * 96
    float* dtm   = x_dbl + (size_t)L_SEQ * (RRANK + 2 * NST);
    float* y     = dtm   + (size_t)L_SEQ * DIN;

    const dim3 blk(256);
    const int NDBL = RRANK + 2 * NST;   // 96

    wmma_gemm_f32<<<dim3(L_SEQ / 64, (2 * DIN + 63) / 64), blk, 0, stream>>>(
        x, W_in, xz, L_SEQ, 2 * DIN, DM, DM, DM, 2 * DIN, nullptr, 0);

    conv_silu_kernel<<<dim3(DIN / 256, L_SEQ), blk, 0, stream>>>(
        xz, conv_w, conv_b, xc);

    wmma_gemm_f32<<<dim3(L_SEQ / 64, (NDBL + 63) / 64), blk, 0, stream>>>(
        xc, W_x, x_dbl, L_SEQ, NDBL, DIN, DIN, DIN, NDBL, nullptr, 0);

    wmma_gemm_f32<<<dim3(L_SEQ / 64, (DIN + 63) / 64), blk, 0, stream>>>(
        x_dbl, W_dt, dtm, L_SEQ, DIN, RRANK, NDBL, RRANK, DIN, b_dt, 1);

    scan_kernel<<<dim3(DIN / 16), blk, 0, stream>>>(
        dtm, x_dbl, xc, xz, A_log, D_param, y);

    wmma_gemm_f32<<<dim3(L_SEQ / 64, (DM + 63) / 64), blk, 0, stream>>>(
        y, W_out, out, L_SEQ, DM, DIN, DIN, DIN, DM, nullptr, 0);
}